// Local_around_edge_loss_68444598829428
// MI455X (gfx1250) — compile-verified
//
#include <hip/hip_runtime.h>
#include <hip/hip_bf16.h>

// MI455X / gfx1250, wave32.
// Bandwidth-bound kernel (~134 MB streamed once, 23.3 TB/s -> ~5.8 us floor).
// One wave per 4x4x4 block; per-block masked probability reduction done on the
// matrix unit with V_WMMA_F32_16X16X4_F32 (fp32-exact), histogram via ballots.

typedef __attribute__((ext_vector_type(2))) float v2f;
typedef __attribute__((ext_vector_type(8))) float v8f;

#define NCLS     12
#define SPATIAL  (64 * 64 * 64)      // 262144
#define NBLOCKS  32768               // 8 * 16^3
#define LSTRIDE  68                  // floats; (4c+lane)%64 distinct -> conflict-free

__global__ __launch_bounds__(256)
void edge_loss_stage1(const float* __restrict__ logits,     // [8,12,64,64,64]
                      const int*   __restrict__ label,      // [8,1,64,64,64]
                      const int*   __restrict__ lweight,    // [8,1,64,64,64]
                      const float* __restrict__ sketch,     // [8,2,64,64,64]
                      float*       __restrict__ blk_res)    // [NBLOCKS][2]
{
    __shared__ float pm_all[8][16][LSTRIDE];                // 34816 B / WG

    const int wave = threadIdx.x >> 5;
    const int lane = threadIdx.x & 31;
    const int n    = blockIdx.x * 8 + wave;                 // block id, one wave each

    const int db = n & 15, wb = (n >> 4) & 15, hb = (n >> 8) & 15, b = n >> 12;
    const int sp_base = (hb * 4) * 4096 + (wb * 4) * 64 + (db * 4);

    float (*P)[LSTRIDE] = pm_all[wave];                     // wave-private slice

    // Zero A-matrix rows 12..15 (classes beyond NCLS must contribute 0).
    for (int t = lane; t < 4 * 64; t += 32) {
        P[12 + (t >> 6)][t & 63] = 0.0f;
    }

    // ---- Stage 1: per-voxel softmax, masked probs -> LDS in [class][voxel] ----
    int lab[2], wm[2], sk[2];
#pragma unroll
    for (int h = 0; h < 2; ++h) {
        const int v = lane + h * 32;                        // voxel 0..63 in block
        const int k = v & 3, j = (v >> 2) & 3, i = v >> 4;
        const int sp = sp_base + i * 4096 + j * 64 + k;

        const int gi = b * SPATIAL + sp;
        lab[h] = label[gi];
        wm[h]  = lweight[gi];
        const float s0 = sketch[(size_t)(b * 2) * SPATIAL + sp];
        const float s1 = sketch[(size_t)(b * 2 + 1) * SPATIAL + sp];
        sk[h] = (s1 > s0) ? 1 : 0;                          // argmax of 2 channels

        const float* lp = logits + (size_t)(b * NCLS) * SPATIAL + sp;
        float x[NCLS];
        float m = -3.402823466e38f;
#pragma unroll
        for (int c = 0; c < NCLS; ++c) { x[c] = lp[(size_t)c * SPATIAL]; m = fmaxf(m, x[c]); }
        float s = 0.0f;
#pragma unroll
        for (int c = 0; c < NCLS; ++c) { x[c] = __expf(x[c] - m); s += x[c]; }
        const float inv = (float)wm[h] / s;                 // fold mask into probs
#pragma unroll
        for (int c = 0; c < NCLS; ++c) P[c][v] = x[c] * inv;
    }

    // Wave-internal LDS RAW fence (lanes are lockstep; just drain DScnt).
    asm volatile("s_wait_dscnt 0" ::: "memory");

    // ---- Block predicates + exact masked label histogram via ballots ----
    const unsigned bw0 = __builtin_amdgcn_ballot_w32(wm[0] != 0);
    const unsigned bw1 = __builtin_amdgcn_ballot_w32(wm[1] != 0);
    const int count_sum = __popc(bw0) + __popc(bw1);
    const bool valid_here = count_sum > 0;
    const bool edge_here  = __builtin_amdgcn_ballot_w32((sk[0] | sk[1]) != 0) != 0u;
    const bool all_lt255  =
        __builtin_amdgcn_ballot_w32((lab[0] < 255) && (lab[1] < 255)) == 0xFFFFFFFFu;

    float cnt[NCLS];
#pragma unroll
    for (int c = 0; c < NCLS; ++c) {
        const unsigned c0 = __builtin_amdgcn_ballot_w32((wm[0] != 0) && (lab[0] == c));
        const unsigned c1 = __builtin_amdgcn_ballot_w32((wm[1] != 0) && (lab[1] == c));
        cnt[c] = (float)(__popc(c0) + __popc(c1));
    }

    // ---- Stage 2: pred[c] = sum_s probs_masked[c,s] via fp32 WMMA, B = ones ----
    // A 16x4 f32 layout: lanes 0-15 hold (M=lane, K=0/1), lanes 16-31 (M=lane-16, K=2/3).
    const int row  = lane & 15;
    const int koff = (lane >> 4) * 2;
    v2f bones; bones.x = 1.0f; bones.y = 1.0f;
    v8f acc = {};
#pragma unroll
    for (int t = 0; t < 16; ++t) {                          // K = 64 in steps of 4
        const int s = t * 4 + koff;
        v2f a;
        a.x = P[row][s];
        a.y = P[row][s + 1];
        acc = __builtin_amdgcn_wmma_f32_16x16x4_f32(
            /*neg_a=*/false, a, /*neg_b=*/false, bones,
            /*c_mod=*/(short)0, acc, /*reuse_a=*/false, /*reuse_b=*/false);
    }
    // D layout: VGPR r -> class r (lanes 0-15) / class 8+r (lanes 16-31); all N cols equal.

    float psum_loc = 0.0f;
#pragma unroll
    for (int r = 0; r < 8; ++r) psum_loc += acc[r];
    const float pred_sum = psum_loc + __shfl_xor(psum_loc, 16, 32);

    const float ps_den = (pred_sum  > 0.0f) ? pred_sum : 1.0f;
    const float cs_den = (float)(count_sum > 0 ? count_sum : 1);

    const int cbase = (lane < 16) ? 0 : 8;
    float klp = 0.0f;
#pragma unroll
    for (int r = 0; r < 8; ++r) {
        const int c = cbase + r;
        if (c < NCLS) {
            const float q = cnt[c] / cs_den;
            const float p = acc[r] / ps_den;
            if (q > 0.0f) {
                const float p_safe = (p > 0.0f) ? p : 1.0f;
                klp += q * (__logf(q) - __logf(p_safe));
            }
        }
    }
    const float kl = klp + __shfl_xor(klp, 16, 32);         // lanes l and l^16 hold complements

    if (lane == 0) {
        const bool ok = edge_here && valid_here && all_lt255;
        blk_res[2 * n]     = ok ? kl : 0.0f;
        blk_res[2 * n + 1] = ok ? 1.0f : 0.0f;
    }
}

__global__ __launch_bounds__(256)
void edge_loss_stage2(const float* __restrict__ blk_res, float* __restrict__ out)
{
    __shared__ float st[256], sn[256];
    float t = 0.0f, c = 0.0f;
    for (int i = threadIdx.x; i < NBLOCKS; i += 256) {
        t += blk_res[2 * i];
        c += blk_res[2 * i + 1];
    }
    st[threadIdx.x] = t; sn[threadIdx.x] = c;
    __syncthreads();
    for (int o = 128; o > 0; o >>= 1) {
        if (threadIdx.x < (unsigned)o) {
            st[threadIdx.x] += st[threadIdx.x + o];
            sn[threadIdx.x] += sn[threadIdx.x + o];
        }
        __syncthreads();
    }
    if (threadIdx.x == 0) out[0] = (sn[0] > 0.0f) ? st[0] / sn[0] : 0.0f;
}

extern "C" void kernel_launch(void* const* d_in, const int* in_sizes, int n_in,
                              void* d_out, int out_size, void* d_ws, size_t ws_size,
                              hipStream_t stream) {
    const float* logits  = (const float*)d_in[0];   // output [8,12,64,64,64] f32
    const int*   label   = (const int*)  d_in[1];   // [8,1,64,64,64] i32
    const int*   lweight = (const int*)  d_in[2];   // [8,1,64,64,64] i32
    const float* sketch  = (const float*)d_in[3];   // [8,2,64,64,64] f32
    float* blk_res = (float*)d_ws;                  // 32768 * 2 floats = 256 KB
    float* out = (float*)d_out;

    edge_loss_stage1<<<NBLOCKS / 8, 256, 0, stream>>>(logits, label, lweight, sketch, blk_res);
    edge_loss_stage2<<<1, 256, 0, stream>>>(blk_res, out);
}